// GateModule_19335942766947
// MI455X (gfx1250) — compile-verified
//
#include <hip/hip_runtime.h>

typedef __attribute__((ext_vector_type(2))) float v2f;
typedef __attribute__((ext_vector_type(8))) float v8f;

#define B_ 4
#define N_ 8
#define T_ 4096
#define D_ 1024

// Block: 256 threads = 8 waves; wave n owns concept n, block owns a 16-token tile.
__global__ __launch_bounds__(256, 2)
void gate_argmax_combine_kernel(const float* __restrict__ e,
                                const float* __restrict__ w,
                                float* __restrict__ e_out,
                                int*   __restrict__ onehot,
                                float* __restrict__ maxval,
                                int*   __restrict__ maxind) {
    __shared__ float lds_w[N_ * D_];   // 32 KB staged scaler weights
    __shared__ float lds_s[N_ * 16];   // raw logits [concept][token]
    __shared__ float lds_wv[16];       // winning sigmoid value per token
    __shared__ int   lds_wi[16];       // winning concept per token

    const int tid = threadIdx.x;
    const int b  = blockIdx.x >> 8;          // T_/16 = 256 tiles per batch
    const int t0 = (blockIdx.x & 255) << 4;

    // Stage all weights in LDS (cooperative, coalesced).
    for (int i = tid; i < N_ * D_; i += 256) lds_w[i] = w[i];
    __syncthreads();

    const int n    = tid >> 5;   // wave id == concept id
    const int lane = tid & 31;
    const int m    = lane & 15;  // token row within tile (A-matrix M)
    const int h    = lane >> 4;  // K half: 0 -> K{0,1}, 1 -> K{2,3}

    // A rows: e[b, n, t0+m, :]; each lane loads its K-pair (8B) per step.
    const float* arow = e + (((size_t)(b * N_ + n) * T_) + (size_t)(t0 + m)) * D_ + 2 * h;
    // B (4x16, all columns identical): w[n, d + K]; VGPR0 = K{0|2}, VGPR1 = K{1|3}.
    const float* brow = lds_w + n * D_ + 2 * h;

    v8f acc = {};
    #pragma unroll 8
    for (int d = 0; d < D_; d += 4) {
        v2f a  = *(const v2f*)(arow + d);
        v2f bb = *(const v2f*)(brow + d);
        // D = A(16x4 f32) * B(4x16 f32) + C, accumulate logits in column 0.
        acc = __builtin_amdgcn_wmma_f32_16x16x4_f32(false, a, false, bb,
                                                    (short)0, acc, false, false);
    }

    // Column N=0: lane 0 holds tokens 0..7 in acc[0..7], lane 16 holds tokens 8..15.
    if (m == 0) {
        #pragma unroll
        for (int r = 0; r < 8; ++r) lds_s[n * 16 + h * 8 + r] = acc[r];
    }
    __syncthreads();

    // Per-token argmax over concepts on raw logits (sigmoid is monotonic;
    // strict > keeps the first max index, matching jnp.argmax).
    if (tid < 16) {
        float best = lds_s[tid];
        int   bi   = 0;
        #pragma unroll
        for (int c = 1; c < N_; ++c) {
            float s = lds_s[c * 16 + tid];
            if (s > best) { best = s; bi = c; }
        }
        float val = 1.0f / (1.0f + __expf(-best));
        lds_wv[tid] = val;
        lds_wi[tid] = bi;
        maxval[b * T_ + t0 + tid] = val;
        maxind[b * T_ + t0 + tid] = bi;
    }
    __syncthreads();

    // One-hot [b, n, t] writes.
    if (tid < 128) {
        int nn = tid >> 4, mm = tid & 15;
        onehot[(size_t)(b * N_ + nn) * T_ + (size_t)(t0 + mm)] = (lds_wi[mm] == nn) ? 1 : 0;
    }

    // Gather winning concept rows, scale, store e_out.
    // 256 threads * float4 = 1024 floats = one token row per iteration (coalesced).
    #pragma unroll 4
    for (int mm = 0; mm < 16; ++mm) {
        const int   wi = lds_wi[mm];
        const float wv = lds_wv[mm];
        const float4* src = (const float4*)(e + (((size_t)(b * N_ + wi) * T_) +
                                                 (size_t)(t0 + mm)) * D_) + tid;
        float4 v = *src;
        v.x *= wv; v.y *= wv; v.z *= wv; v.w *= wv;
        *((float4*)(e_out + ((size_t)(b * T_) + (size_t)(t0 + mm)) * D_) + tid) = v;
    }
}

extern "C" void kernel_launch(void* const* d_in, const int* in_sizes, int n_in,
                              void* d_out, int out_size, void* d_ws, size_t ws_size,
                              hipStream_t stream) {
    (void)in_sizes; (void)n_in; (void)out_size; (void)d_ws; (void)ws_size;

    const float* e = (const float*)d_in[0];   // [B, N, T, D] fp32
    const float* w = (const float*)d_in[1];   // [N, D, 1]    fp32

    float* out    = (float*)d_out;
    float* e_out  = out;                                        // [B, T, D]
    size_t off    = (size_t)B_ * T_ * D_;
    int*   onehot = (int*)(out + off);                          // [B, N, T, 1] int32
    off          += (size_t)B_ * N_ * T_;
    float* maxval = out + off;                                  // [B, 1, T, 1] fp32
    off          += (size_t)B_ * T_;
    int*   maxind = (int*)(out + off);                          // [B, 1, T, 1] int32

    dim3 grid(B_ * (T_ / 16));
    dim3 block(256);
    gate_argmax_combine_kernel<<<grid, block, 0, stream>>>(e, w, e_out, onehot,
                                                           maxval, maxind);
}